// Attn1_44281112822484
// MI455X (gfx1250) — compile-verified
//
#include <hip/hip_runtime.h>

#define L_DIM 2048
#define B_DIM 128
#define H_DIM 300
#define H_PAD 304   // pad h-dim to multiple of 16 for tiling / float4 alignment

typedef __attribute__((ext_vector_type(2))) float v2f;
typedef __attribute__((ext_vector_type(4))) float v4f;
typedef __attribute__((ext_vector_type(8))) float v8f;

// ---------------------------------------------------------------------------
// u[b][h] = sum_k W[k][h] * hidden[k][b]   (Wt @ hidden), f32 WMMA 16x16x4.
// One wave (32 lanes) per 16x16 output tile. grid = (19 h-tiles, 8 b-tiles).
// A-matrix 16x4 f32 layout: lanes 0-15 -> M=0..15 holding K=0,1 in 2 VGPRs;
// lanes 16-31 -> same M, K=2,3. B-matrix 4x16 mirrored over N. C/D: 8 VGPRs,
// lanes 0-15 N=lane M=r, lanes 16-31 N=lane-16 M=r+8.
// Out-of-range h rows (300..303, last tile only) are handled branchlessly:
// load from a clamped (valid) address, then v_cndmask the value to zero, so
// the WMMA loop carries no exec-mask control flow.
// ---------------------------------------------------------------------------
__global__ __launch_bounds__(32) void k_project(const float* __restrict__ W,
                                                const float* __restrict__ hidden,
                                                float* __restrict__ u) {
  const int tile_m = blockIdx.x;        // h tile: 0..18
  const int tile_n = blockIdx.y;        // b tile: 0..7
  const int lane = threadIdx.x;         // 0..31 (wave32)
  const int mn   = lane & 15;           // M index for A, N index for B
  const int kb   = (lane >> 4) << 1;    // K base: 0 or 2
  const int hg   = tile_m * 16 + mn;    // global h (row of u)
  const int bg   = tile_n * 16 + mn;    // global b (col of u)
  const bool hok = (hg < H_DIM);
  const int hgc  = hok ? hg : (H_DIM - 1);   // clamped: always a legal address

  const float* __restrict__ Wc = W + hgc;          // column hgc of W
  const float* __restrict__ Hc = hidden + bg;      // column bg of hidden

  v8f acc = {};
  for (int k0 = 0; k0 < H_DIM; k0 += 4) {     // 300 % 4 == 0 -> 75 WMMAs
    float a0 = Wc[(k0 + kb) * H_DIM];
    float a1 = Wc[(k0 + kb + 1) * H_DIM];
    v2f av, bv;
    av.x = hok ? a0 : 0.0f;                   // branchless v_cndmask
    av.y = hok ? a1 : 0.0f;
    bv.x = Hc[(k0 + kb) * B_DIM];
    bv.y = Hc[(k0 + kb + 1) * B_DIM];
    acc = __builtin_amdgcn_wmma_f32_16x16x4_f32(
        /*neg_a=*/false, av, /*neg_b=*/false, bv,
        /*c_mod=*/(short)0, acc, /*reuse_a=*/false, /*reuse_b=*/false);
  }

  const int mofs = (lane >> 4) * 8;
#pragma unroll
  for (int r = 0; r < 8; ++r) {
    int h = tile_m * 16 + mofs + r;
    if (h < H_DIM) u[(size_t)(tile_n * 16 + mn) * H_PAD + h] = acc[r];
  }
}

// c[b] = sum_k bias[k] * hidden[k][b]  (tiny)
__global__ __launch_bounds__(128) void k_bias(const float* __restrict__ bias,
                                              const float* __restrict__ hidden,
                                              float* __restrict__ c) {
  const int b = threadIdx.x;
  float s = 0.0f;
  for (int k = 0; k < H_DIM; ++k) s = fmaf(bias[k], hidden[k * B_DIM + b], s);
  c[b] = s;
}

// ---------------------------------------------------------------------------
// energies[b][l] = sum_h q[l][b][h] * u[b][h] + c[b]
// One wave per (l,b) pair; lanes cover h with float4 loads (coalesced 512B
// clauses per wave), xor-shuffle reduce within the wave. q (315MB) is streamed
// once with non-temporal loads so it doesn't evict the hot u rows (150KB,
// L2/WGP$-resident) or the energies written for the softmax pass.
// ---------------------------------------------------------------------------
__global__ __launch_bounds__(256) void k_energy(const float* __restrict__ q,
                                                const float* __restrict__ u,
                                                const float* __restrict__ c,
                                                float* __restrict__ e) {
  const int lane = threadIdx.x & 31;
  const int wave = (blockIdx.x << 3) | (threadIdx.x >> 5);  // 8 waves/block
  const int b = wave & (B_DIM - 1);
  const int l = wave >> 7;

  const v4f* __restrict__ qv =
      (const v4f*)(q + ((size_t)l * B_DIM + b) * H_DIM);  // 1200B row, 16B aligned
  const v4f* __restrict__ uv = (const v4f*)(u + (size_t)b * H_PAD);

  float s = 0.0f;
#pragma unroll
  for (int j = 0; j < 3; ++j) {
    int h4 = j * 32 + lane;           // float4 index; 75 float4 = 300 floats
    if (h4 < H_DIM / 4) {
      v4f qq = __builtin_nontemporal_load(&qv[h4]);  // streaming, NT hint
      v4f uu = uv[h4];                               // reused, keep cached
      s = fmaf(qq.x, uu.x, s);
      s = fmaf(qq.y, uu.y, s);
      s = fmaf(qq.z, uu.z, s);
      s = fmaf(qq.w, uu.w, s);
    }
  }
#pragma unroll
  for (int off = 16; off > 0; off >>= 1) s += __shfl_xor(s, off, 32);
  if (lane == 0) e[(size_t)b * L_DIM + l] = s + c[b];
}

// ---------------------------------------------------------------------------
// In-place softmax over l for each b. 256 threads/row, 8 values/thread in regs.
// ---------------------------------------------------------------------------
__global__ __launch_bounds__(256) void k_softmax(float* __restrict__ e) {
  const int b = blockIdx.x;
  const int tid = threadIdx.x;
  float* __restrict__ row = e + (size_t)b * L_DIM;

  float v[8];
  float mx = -3.402823466e38f;
#pragma unroll
  for (int i = 0; i < 8; ++i) {
    v[i] = row[tid + i * 256];
    mx = fmaxf(mx, v[i]);
  }
#pragma unroll
  for (int off = 16; off > 0; off >>= 1) mx = fmaxf(mx, __shfl_xor(mx, off, 32));

  __shared__ float wmax[8];
  __shared__ float wsum[8];
  if ((tid & 31) == 0) wmax[tid >> 5] = mx;
  __syncthreads();
  float bm = wmax[0];
#pragma unroll
  for (int i = 1; i < 8; ++i) bm = fmaxf(bm, wmax[i]);

  float s = 0.0f;
#pragma unroll
  for (int i = 0; i < 8; ++i) {
    v[i] = expf(v[i] - bm);
    s += v[i];
  }
#pragma unroll
  for (int off = 16; off > 0; off >>= 1) s += __shfl_xor(s, off, 32);
  if ((tid & 31) == 0) wsum[tid >> 5] = s;
  __syncthreads();
  float tot = 0.0f;
#pragma unroll
  for (int i = 0; i < 8; ++i) tot += wsum[i];
  const float inv = 1.0f / tot;
#pragma unroll
  for (int i = 0; i < 8; ++i) row[tid + i * 256] = v[i] * inv;
}

extern "C" void kernel_launch(void* const* d_in, const int* in_sizes, int n_in,
                              void* d_out, int out_size, void* d_ws, size_t ws_size,
                              hipStream_t stream) {
  (void)in_sizes; (void)n_in; (void)out_size; (void)ws_size;
  const float* hidden = (const float*)d_in[0];  // [H, B]
  const float* q      = (const float*)d_in[1];  // [L, B, H]
  const float* W      = (const float*)d_in[2];  // [H, H]
  const float* bias   = (const float*)d_in[3];  // [H]
  float* out = (float*)d_out;                   // [1, B, L] == [B][L] flat

  float* u = (float*)d_ws;                      // [B][H_PAD] = 152 KiB
  float* c = u + (size_t)B_DIM * H_PAD;         // [B]

  k_project<<<dim3(19, 8), 32, 0, stream>>>(W, hidden, u);
  k_bias<<<1, 128, 0, stream>>>(bias, hidden, c);
  k_energy<<<(L_DIM * B_DIM) / 8, 256, 0, stream>>>(q, u, c, out);
  k_softmax<<<B_DIM, 256, 0, stream>>>(out);
}